// CrossAndCompress_71734543778591
// MI455X (gfx1250) — compile-verified
//
#include <hip/hip_runtime.h>

typedef float v2f __attribute__((ext_vector_type(2)));
typedef float v4f __attribute__((ext_vector_type(4)));
typedef float v8f __attribute__((ext_vector_type(8)));

#define BATCH 16384
#define DDIM  1024
#define TILE  16             // rows per block (= WMMA N)
#define NWAVE 8              // 256 threads, wave32
#define KSPAN (DDIM / NWAVE) // 128 K-values per wave
#define THSTR 1028           // padded LDS row stride (banks = (4m + k) % 64)

__global__ __launch_bounds__(256)
void cross_compress_kernel(const float* __restrict__ enc_user,
                           const float* __restrict__ enc_item,
                           const float* __restrict__ t_vv,
                           const float* __restrict__ t_ev,
                           const float* __restrict__ t_ve,
                           const float* __restrict__ t_ee,
                           const float* __restrict__ beta_v,
                           const float* __restrict__ beta_e,
                           float* __restrict__ v_out,
                           float* __restrict__ e_out)
{
    // rows 0..3 = theta_vv/ev/ve/ee, row 4 = zeros (branch-free A for m>=4)
    __shared__ float th_lds[5 * THSTR];
    __shared__ float red[2][NWAVE][4][16]; // per-wave partial D tiles (u/i)
    __shared__ float dots[2][4][16];       // final [mat][theta][row]

    const int tid  = threadIdx.x;
    const int wave = tid >> 5;
    const int lane = tid & 31;
    const int g    = lane >> 4;   // half-lane group -> K sub-pair select
    const int sel  = lane & 15;   // A row (m) and B column (n = tile row)
    const int row0 = blockIdx.x * TILE;

    // ---------- stage thetas into LDS (one-time, tiny) ----------
    {
        const int c = tid * 4;    // 256 threads x 4 floats = 1024 per row
        *(v4f*)&th_lds[0 * THSTR + c] = *(const v4f*)(t_vv + c);
        *(v4f*)&th_lds[1 * THSTR + c] = *(const v4f*)(t_ev + c);
        *(v4f*)&th_lds[2 * THSTR + c] = *(const v4f*)(t_ve + c);
        *(v4f*)&th_lds[3 * THSTR + c] = *(const v4f*)(t_ee + c);
        v4f z = {0.f, 0.f, 0.f, 0.f};
        *(v4f*)&th_lds[4 * THSTR + c] = z;
    }
    __syncthreads();

    // ---------- phase 1: 16x4 dot block via V_WMMA_F32_16X16X4_F32 ----------
    // A(16x4): rows 0..3 = theta K-slices, rows 4..15 read the LDS zero row
    // B(4x16): column n = enc row (row0+n) K-slice
    // D[m][n] += sum_k theta_m[k] * enc[row0+n][k]
    const int arow = (sel < 4) ? sel : 4;            // branch-free A select
    const float* aptr = &th_lds[arow * THSTR];

    v8f acc_u = {0.f,0.f,0.f,0.f,0.f,0.f,0.f,0.f};
    v8f acc_i = {0.f,0.f,0.f,0.f,0.f,0.f,0.f,0.f};

    const float* urow = enc_user + (size_t)(row0 + sel) * DDIM;
    const float* irow = enc_item + (size_t)(row0 + sel) * DDIM;

    #pragma unroll 4
    for (int s = 0; s < KSPAN / 4; ++s) {
        const int kk = wave * KSPAN + s * 4 + g * 2;
        v2f a  = *(const v2f*)(aptr + kk);           // ds_load_b64, no divergence
        v2f bu = *(const v2f*)(urow + kk);
        v2f bi = *(const v2f*)(irow + kk);
        acc_u = __builtin_amdgcn_wmma_f32_16x16x4_f32(
            false, a, false, bu, (short)0, acc_u, false, false);
        acc_i = __builtin_amdgcn_wmma_f32_16x16x4_f32(
            false, a, false, bi, (short)0, acc_i, false, false);
    }

    // ---------- cross-wave reduction (only D rows m=0..3 are live) ----------
    if (lane < 16) {
        #pragma unroll
        for (int m = 0; m < 4; ++m) {
            red[0][wave][m][lane] = acc_u[m];
            red[1][wave][m][lane] = acc_i[m];
        }
    }
    __syncthreads();
    if (tid < 128) {
        const int mat = tid >> 6, m = (tid >> 4) & 3, n = tid & 15;
        float ssum = 0.f;
        #pragma unroll
        for (int w2 = 0; w2 < NWAVE; ++w2) ssum += red[mat][w2][m][n];
        dots[mat][m][n] = ssum;
    }
    __syncthreads();

    // ---------- phase 2: streaming epilogue (HBM-bound; re-reads hit L2) ----------
    #pragma unroll
    for (int rr = 0; rr < 2; ++rr) {
        const int r = wave * 2 + rr;                 // 8 waves x 2 rows = 16
        const size_t row = (size_t)(row0 + r);
        const float e_vv = dots[1][0][r];            // enc_item . t_vv
        const float v_ev = dots[0][1][r];            // enc_user . t_ev
        const float e_ve = dots[1][2][r];            // enc_item . t_ve
        const float v_ee = dots[0][3][r];            // enc_user . t_ee
        const float* up = enc_user + row * DDIM;
        const float* ip = enc_item + row * DDIM;
        float* vp = v_out + row * DDIM;
        float* ep = e_out + row * DDIM;
        #pragma unroll 2
        for (int base = 0; base < DDIM; base += 128) {
            const int col = base + lane * 4;
            v4f u  = *(const v4f*)(up + col);
            v4f it = *(const v4f*)(ip + col);
            v4f bv = *(const v4f*)(beta_v + col);
            v4f be = *(const v4f*)(beta_e + col);
            v4f vo = u * e_vv + it * v_ev + bv;
            v4f eo = u * e_ve + it * v_ee + be;
            __builtin_nontemporal_store(vo, (v4f*)(vp + col));
            __builtin_nontemporal_store(eo, (v4f*)(ep + col));
        }
    }
}

extern "C" void kernel_launch(void* const* d_in, const int* in_sizes, int n_in,
                              void* d_out, int out_size, void* d_ws, size_t ws_size,
                              hipStream_t stream) {
    const float* enc_user = (const float*)d_in[0];
    const float* enc_item = (const float*)d_in[1];
    const float* t_vv     = (const float*)d_in[2];
    const float* t_ev     = (const float*)d_in[3];
    const float* t_ve     = (const float*)d_in[4];
    const float* t_ee     = (const float*)d_in[5];
    const float* beta_v   = (const float*)d_in[6];
    const float* beta_e   = (const float*)d_in[7];
    float* v_out = (float*)d_out;
    float* e_out = (float*)d_out + (size_t)BATCH * DDIM;

    dim3 grid(BATCH / TILE);
    dim3 block(256);
    hipLaunchKernelGGL(cross_compress_kernel, grid, block, 0, stream,
                       enc_user, enc_item, t_vv, t_ev, t_ve, t_ee,
                       beta_v, beta_e, v_out, e_out);
}